// CTC_34153579938505
// MI455X (gfx1250) — compile-verified
//
#include <hip/hip_runtime.h>

#define T_DIM 1024
#define B_DIM 128
#define C_DIM 256
#define S_DIM 128
#define L_DIM 257            // 2*S + 1
#define NEGV  (-1.0e9f)
#define LOG2E 1.4426950408889634f
#define LN2   0.6931471805599453f

#define AS1 __attribute__((address_space(1)))
#define AS3 __attribute__((address_space(3)))

typedef int v4i __attribute__((ext_vector_type(4)));

#if __has_builtin(__builtin_amdgcn_global_load_async_to_lds_b128) && \
    __has_builtin(__builtin_amdgcn_s_wait_asynccnt)
#define USE_ASYNC 1
#else
#define USE_ASYNC 0
#if defined(__AMDGCN__)
#warning "gfx1250 async-to-LDS builtins not found; using synchronous staging fallback"
#endif
#endif

// Generic->LDS pointer: ISA 10.2 — generic LDS address low 32 bits are the LDS byte offset.
__device__ __forceinline__ AS3 v4i* lds_cast4(void* p) {
  return (AS3 v4i*)(unsigned int)(unsigned long long)p;
}
// Generic->global pointer: identical bit pattern.
__device__ __forceinline__ AS1 v4i* gbl_cast4(const void* p) {
  return (AS1 v4i*)(unsigned long long)p;
}

// fast base-2 exp/log (single v_exp_f32 / v_log_f32)
__device__ __forceinline__ float fexp2(float x) { return __builtin_amdgcn_exp2f(x); }
__device__ __forceinline__ float flog2(float x) { return __builtin_amdgcn_logf(x); }

__global__ __launch_bounds__(288) void ctc_alpha_kernel(
    const float* __restrict__ lp,     // [T, B, C] log-probs
    const int*   __restrict__ y,      // [B, S]
    const int*   __restrict__ ilen,   // [B]
    const int*   __restrict__ tlen,   // [B]
    float*       __restrict__ per_b)  // [B] per-sample loss / target_len
{
  __shared__ float abuf[2][L_DIM + 2];   // ping-pong alpha (log2 units), 2 NEG pad slots
  __shared__ float lprow[4][C_DIM];      // 4-deep staged log-prob rows (natural log units)
  __shared__ int   lab[L_DIM];
  __shared__ int   skiplds[L_DIM];

  const int b   = blockIdx.x;
  const int tid = threadIdx.x;
  const int Tin = ilen[b];
  const float* lpb = lp + (size_t)b * C_DIM;  // row t lives at lpb + t*B*C

  // Extended label sequence + skip-allowed flags (constant over t).
  if (tid < L_DIM) {
    int l = 0, sk = 0;
    if (tid & 1) {
      l = y[b * S_DIM + (tid >> 1)];
      if (tid >= 3) sk = (l != y[b * S_DIM + (tid >> 1) - 1]) ? 1 : 0;
    }
    lab[tid]     = l;
    skiplds[tid] = sk;
  }
  if (tid < 2) { abuf[0][tid] = NEGV; abuf[1][tid] = NEGV; }

  // Stage row 0 into lprow[0] and wait (needed for alpha0 init).
#if USE_ASYNC
  if (tid < 64) {
    __builtin_amdgcn_global_load_async_to_lds_b128(
        gbl_cast4(lpb + tid * 4), lds_cast4(&lprow[0][tid * 4]), 0, 0);
  }
  __builtin_amdgcn_s_wait_asynccnt(0);
#else
  if (tid < C_DIM) lprow[0][tid] = lpb[tid];
#endif
  __syncthreads();

  int myLab = 0, mySkip = 0;
  if (tid < L_DIM) { myLab = lab[tid]; mySkip = skiplds[tid]; }

  // alpha at t = 0 (log2 units): only s=0 (blank) and s=1 (first label) reachable.
  if (tid < L_DIM) {
    float v = NEGV;
    if (tid == 0)      v = lprow[0][0]     * LOG2E;
    else if (tid == 1) v = lprow[0][myLab] * LOG2E;
    abuf[0][tid + 2] = v;
  }

  // Prefetch rows 1..3 (distance-3 pipeline -> per-wave ASYNCcnt = 3).
#if USE_ASYNC
  if (tid < 64) {
    for (int r = 1; r <= 3; ++r) {
      __builtin_amdgcn_global_load_async_to_lds_b128(
          gbl_cast4(lpb + (size_t)r * B_DIM * C_DIM + tid * 4),
          lds_cast4(&lprow[r][tid * 4]), 0, 0);
    }
  }
#else
  if (tid < C_DIM) {
    for (int r = 1; r <= 3; ++r)
      lprow[r][tid] = lpb[(size_t)r * B_DIM * C_DIM + tid];
  }
#endif

#pragma unroll 4
  for (int t = 1; t < T_DIM; ++t) {
#if USE_ASYNC
    // Retire only the oldest transfer (row t); rows t+1, t+2 stay in flight.
    __builtin_amdgcn_s_wait_asynccnt(2);
#endif
    __syncthreads();                       // row t staged by all waves; alpha_{t-1} visible

    const int tp3 = t + 3;
    if (tp3 < T_DIM) {                     // keep pipeline 3 rows deep
#if USE_ASYNC
      if (tid < 64) {
        __builtin_amdgcn_global_load_async_to_lds_b128(
            gbl_cast4(lpb + (size_t)tp3 * B_DIM * C_DIM + tid * 4),
            lds_cast4(&lprow[tp3 & 3][tid * 4]), 0, 0);
      }
#else
      if (tid < C_DIM) lprow[tp3 & 3][tid] = lpb[(size_t)tp3 * B_DIM * C_DIM + tid];
#endif
    }

    if (tid < L_DIM) {
      const float* ap = abuf[(t - 1) & 1];
      float a0 = ap[tid + 2];                       // alpha[s]     (log2 units)
      float a1 = ap[tid + 1];                       // alpha[s-1]   (pad => NEG)
      float a2 = mySkip ? ap[tid] : NEGV;           // alpha[s-2] if skip allowed
      float m  = fmaxf(a0, fmaxf(a1, a2));          // v_max3
      float sum = fexp2(a0 - m) + fexp2(a1 - m) + fexp2(a2 - m);
      float nv  = m + flog2(sum) + lprow[t & 3][myLab] * LOG2E;
      abuf[t & 1][tid + 2] = (t < Tin) ? nv : a0;   // freeze past input length
    }
  }

  __syncthreads();
  if (tid == 0) {
    const float* aT = abuf[(T_DIM - 1) & 1];
    const int tl = tlen[b];
    float a_hi = aT[2 * tl + 2];       // alpha[2*tl]     (log2 units)
    float a_lo = aT[2 * tl + 1];       // alpha[2*tl - 1]
    float m   = fmaxf(a_hi, a_lo);
    float ll2 = m + flog2(fexp2(a_hi - m) + fexp2(a_lo - m));
    float loss = -ll2 * LN2;           // back to natural-log units
    if (!(loss < 0.5e9f)) loss = 0.0f; // zero_infinity
    per_b[b] = loss / (float)tl;       // reduction='mean' divides by target length
  }
}

__global__ __launch_bounds__(128) void ctc_reduce_kernel(const float* __restrict__ per_b,
                                                         float* __restrict__ out)
{
  __shared__ float sm[B_DIM];
  const int tid = threadIdx.x;
  sm[tid] = per_b[tid];
  __syncthreads();
  for (int off = B_DIM / 2; off > 0; off >>= 1) {
    if (tid < off) sm[tid] += sm[tid + off];
    __syncthreads();
  }
  if (tid == 0) out[0] = sm[0] * (1.0f / (float)B_DIM);
}

extern "C" void kernel_launch(void* const* d_in, const int* in_sizes, int n_in,
                              void* d_out, int out_size, void* d_ws, size_t ws_size,
                              hipStream_t stream) {
  (void)in_sizes; (void)n_in; (void)out_size; (void)ws_size;
  const float* lp = (const float*)d_in[0];
  const int*   y  = (const int*)d_in[1];
  const int*   il = (const int*)d_in[2];
  const int*   tl = (const int*)d_in[3];
  float* per_b = (float*)d_ws;   // 128 floats of scratch

  ctc_alpha_kernel<<<B_DIM, 288, 0, stream>>>(lp, y, il, tl, per_b);
  ctc_reduce_kernel<<<1, B_DIM, 0, stream>>>(per_b, (float*)d_out);
}